// SS_GAT_20358144983282
// MI455X (gfx1250) — compile-verified
//
#include <hip/hip_runtime.h>
#include <hip/hip_bf16.h>

typedef float v2f __attribute__((ext_vector_type(2)));
typedef float v8f __attribute__((ext_vector_type(8)));

#define NNODES 50000
#define NEDGES 800000
#define ETOT   (NEDGES + NNODES)
#define NGRAPH 128
#define INCH   256
#define HIDC   64
#define NHEADS 4
#define OUTC   16

// ---------------------------------------------------------------------------
// CDNA5 async memory->LDS copy (ASYNCcnt-tracked), per CDNA5 ISA.
// lds_off = 32-bit LDS byte address (low 32 bits of the generic pointer),
// src     = 64-bit global address in a VGPR pair.
// ---------------------------------------------------------------------------
__device__ __forceinline__ void async_copy_b128(unsigned lds_off, const float* src) {
  asm volatile("global_load_async_to_lds_b128 %0, %1, off"
               :: "v"(lds_off), "v"(src) : "memory");
}
__device__ __forceinline__ void wait_asynccnt0() {
  asm volatile("s_wait_asynccnt 0x0" ::: "memory");
}

// ---------------------------------------------------------------------------
// f32 WMMA GEMM: C[MxN] = A[MxK] * B[KxN], all row-major f32.
// Block = 256 threads (8 waves). Block tile 128(M) x 64(N). K staged 16 deep.
// Tiles staged with GLOBAL_LOAD_ASYNC_TO_LDS_B128 (no VGPR round-trip).
// Each wave owns a 32x32 C sub-tile = 2x2 grid of 16x16 WMMA tiles.
// ---------------------------------------------------------------------------
__global__ __launch_bounds__(256)
void wmma_gemm_f32_kernel(const float* __restrict__ A, const float* __restrict__ B,
                          float* __restrict__ C, int M, int N, int K) {
  __shared__ float As[128][20];   // [m][k] (16 used); pad 20 -> rows 16B aligned,
                                  // 16-lane frag reads hit 16 distinct banks
  __shared__ float Bs[16][68];    // [k][n]; pad 68 -> rows 16B aligned

  const int tid  = threadIdx.x;
  const int lane = tid & 31;
  const int wave = tid >> 5;
  const int wm   = wave & 3;      // 4 wave-rows  -> 32 rows each
  const int wn   = wave >> 2;     // 2 wave-cols  -> 32 cols each
  const int m0   = blockIdx.x * 128;
  const int n0   = blockIdx.y * 64;
  const int half = lane >> 4;     // K-half select per ISA A/B layout
  const int lm   = lane & 15;

  // Zero the tail rows once; async staging never touches them, so they stay 0.
  if (tid < 128 && (m0 + tid) >= M) {
#pragma unroll
    for (int kk = 0; kk < 16; ++kk) As[tid][kk] = 0.0f;
  }

  const v8f vzero = {0.f, 0.f, 0.f, 0.f, 0.f, 0.f, 0.f, 0.f};
  v8f acc[2][2];
#pragma unroll
  for (int i = 0; i < 2; ++i)
#pragma unroll
    for (int j = 0; j < 2; ++j) acc[i][j] = vzero;

  for (int k0 = 0; k0 < K; k0 += 16) {
    // Stage A block: 128 rows x 16 k = 512 x b128 chunks, 2 per thread
#pragma unroll
    for (int j = 0; j < 2; ++j) {
      int i  = tid + j * 256;               // 0..511
      int mm = i >> 2, kc = (i & 3) * 4;
      int gr = m0 + mm;
      if (gr < M) {
        unsigned dst = (unsigned)(uintptr_t)&As[mm][kc];
        async_copy_b128(dst, A + (size_t)gr * K + (k0 + kc));
      }
    }
    // Stage B block: 16 k x 64 n = 256 x b128 chunks, 1 per thread
    {
      int kk = tid >> 4, nc = (tid & 15) * 4;
      unsigned dst = (unsigned)(uintptr_t)&Bs[kk][nc];
      async_copy_b128(dst, B + (size_t)(k0 + kk) * N + (n0 + nc));
    }
    wait_asynccnt0();
    __syncthreads();

#pragma unroll
    for (int kk = 0; kk < 16; kk += 4) {
      v2f afrag[2], bfrag[2];
#pragma unroll
      for (int mt = 0; mt < 2; ++mt) {
        int row = wm * 32 + mt * 16 + lm;   // A: lanes 0-15 K={0,1}, 16-31 K={2,3}
        afrag[mt].x = As[row][kk + 2 * half];
        afrag[mt].y = As[row][kk + 2 * half + 1];
      }
#pragma unroll
      for (int nt = 0; nt < 2; ++nt) {
        int col = wn * 32 + nt * 16 + lm;
        bfrag[nt].x = Bs[kk + 2 * half][col];
        bfrag[nt].y = Bs[kk + 2 * half + 1][col];
      }
#pragma unroll
      for (int mt = 0; mt < 2; ++mt)
#pragma unroll
        for (int nt = 0; nt < 2; ++nt)
          acc[mt][nt] = __builtin_amdgcn_wmma_f32_16x16x4_f32(
              false, afrag[mt], false, bfrag[nt], (short)0, acc[mt][nt],
              false, false);
    }
    __syncthreads();
  }

  // Store C: VGPR r of the 16x16 f32 D tile holds rows r (lanes 0-15) / r+8 (16-31)
#pragma unroll
  for (int mt = 0; mt < 2; ++mt) {
#pragma unroll
    for (int nt = 0; nt < 2; ++nt) {
      int col = n0 + wn * 32 + nt * 16 + lm;
#pragma unroll
      for (int r = 0; r < 8; ++r) {
        int row = m0 + wm * 32 + mt * 16 + half * 8 + r;
        if (row < M) C[(size_t)row * N + col] = acc[mt][nt][r];
      }
    }
  }
}

// ---------------------------------------------------------------------------
// Utility / element-wise kernels
// ---------------------------------------------------------------------------
__global__ void fill_f32(float* __restrict__ p, float v, int n) {
  int i = blockIdx.x * blockDim.x + threadIdx.x;
  if (i < n) p[i] = v;
}

// a_src[n,h] = sum_c H[n, h*64+c]*att_src[h,c] ; a_dst likewise
__global__ void att_reduce_kernel(const float* __restrict__ H,
                                  const float* __restrict__ atts,
                                  const float* __restrict__ attd,
                                  float* __restrict__ osrc, float* __restrict__ odst,
                                  int heads, int nNodes) {
  int i = blockIdx.x * blockDim.x + threadIdx.x;
  if (i >= nNodes * heads) return;
  int n = i / heads, h = i % heads;
  const float* hp = H + ((size_t)n * heads + h) * HIDC;
  const float* as = atts + h * HIDC;
  const float* ad = attd + h * HIDC;
  float s = 0.f, d = 0.f;
#pragma unroll 8
  for (int c = 0; c < HIDC; ++c) {
    float v = hp[c];
    s += v * as[c];
    d += v * ad[c];
  }
  osrc[i] = s;
  odst[i] = d;
}

// ordered-uint encode for float atomicMax (handles negatives)
__device__ __forceinline__ unsigned f2ord(float f) {
  unsigned u = __float_as_uint(f);
  return (u & 0x80000000u) ? ~u : (u | 0x80000000u);
}
__device__ __forceinline__ float ord2f(unsigned u) {
  return (u & 0x80000000u) ? __uint_as_float(u & 0x7FFFFFFFu)
                           : __uint_as_float(~u);
}

__global__ void edge_max_kernel(const int* __restrict__ ei,
                                const float* __restrict__ asrc,
                                const float* __restrict__ adst,
                                unsigned* __restrict__ mx, int heads) {
  int e = blockIdx.x * blockDim.x + threadIdx.x;
  if (e >= ETOT) return;
  int s = (e < NEDGES) ? ei[e] : (e - NEDGES);
  int d = (e < NEDGES) ? ei[NEDGES + e] : (e - NEDGES);
  for (int h = 0; h < heads; ++h) {
    float v = asrc[s * heads + h] + adst[d * heads + h];
    v = (v > 0.f) ? v : 0.2f * v;          // leaky_relu 0.2
    atomicMax(&mx[d * heads + h], f2ord(v));
  }
}

__global__ void edge_exp_kernel(const int* __restrict__ ei,
                                const float* __restrict__ asrc,
                                const float* __restrict__ adst,
                                const unsigned* __restrict__ mx,
                                float* __restrict__ den,
                                float* __restrict__ eexp, int heads) {
  int e = blockIdx.x * blockDim.x + threadIdx.x;
  if (e >= ETOT) return;
  int s = (e < NEDGES) ? ei[e] : (e - NEDGES);
  int d = (e < NEDGES) ? ei[NEDGES + e] : (e - NEDGES);
  for (int h = 0; h < heads; ++h) {
    float v = asrc[s * heads + h] + adst[d * heads + h];
    v = (v > 0.f) ? v : 0.2f * v;
    float ex = expf(v - ord2f(mx[d * heads + h]));
    eexp[(size_t)e * heads + h] = ex;
    atomicAdd(&den[d * heads + h], ex);
  }
}

// 64 threads per edge (channel dim); scatter alpha * H[src] into acc[dst].
__global__ __launch_bounds__(256)
void edge_msg_kernel(const int* __restrict__ ei,
                     const float* __restrict__ eexp,
                     const float* __restrict__ den,
                     const float* __restrict__ H,
                     float* __restrict__ acc, int heads) {
  int t = threadIdx.x;
  int e = blockIdx.x * 4 + (t >> 6);
  int c = t & 63;
  if (e >= ETOT) return;
  int s = (e < NEDGES) ? ei[e] : (e - NEDGES);
  int d = (e < NEDGES) ? ei[NEDGES + e] : (e - NEDGES);
  const float* hsrc = H + (size_t)s * heads * HIDC;
  float* adst = acc + (size_t)d * heads * HIDC;
  __builtin_prefetch(hsrc + c, 0, 1);      // global_prefetch_b8 on the L2-hot row
  for (int h = 0; h < heads; ++h) {
    float alpha = eexp[(size_t)e * heads + h] / (den[d * heads + h] + 1e-16f);
    atomicAdd(&adst[h * HIDC + c], alpha * hsrc[h * HIDC + c]);
  }
}

__global__ void bias_relu_kernel(float* __restrict__ x, const float* __restrict__ b,
                                 int F, int n) {
  int i = blockIdx.x * blockDim.x + threadIdx.x;
  if (i < n) x[i] = fmaxf(x[i] + b[i % F], 0.f);
}

__global__ void pool_kernel(const float* __restrict__ h, const int* __restrict__ batch,
                            float* __restrict__ pool, float* __restrict__ cnt) {
  int i = blockIdx.x * blockDim.x + threadIdx.x;
  if (i >= NNODES * HIDC) return;
  int n = i >> 6, c = i & 63;
  int g = batch[n];
  atomicAdd(&pool[g * HIDC + c], h[i]);
  if (c == 0) atomicAdd(&cnt[g], 1.0f);
}

// One thread per graph: mean -> Linear(64,32)+ReLU -> Linear(32,16) -> log_softmax
__global__ __launch_bounds__(128)
void classifier_kernel(const float* __restrict__ pool, const float* __restrict__ cnt,
                       const float* __restrict__ w1, const float* __restrict__ b1,
                       const float* __restrict__ w2, const float* __restrict__ b2,
                       float* __restrict__ out) {
  int g = threadIdx.x;
  if (g >= NGRAPH) return;
  float inv = 1.0f / fmaxf(cnt[g], 1.0f);
  float pooled[HIDC];
#pragma unroll 8
  for (int j = 0; j < HIDC; ++j) pooled[j] = pool[g * HIDC + j] * inv;
  float t1[HIDC / 2];
  for (int i = 0; i < HIDC / 2; ++i) {
    float s = b1[i];
#pragma unroll 8
    for (int j = 0; j < HIDC; ++j) s += pooled[j] * w1[j * (HIDC / 2) + i];
    t1[i] = fmaxf(s, 0.f);
  }
  float lg[OUTC];
  float mx = -3.4e38f;
  for (int k = 0; k < OUTC; ++k) {
    float s = b2[k];
#pragma unroll 8
    for (int i = 0; i < HIDC / 2; ++i) s += t1[i] * w2[i * OUTC + k];
    lg[k] = s;
    mx = fmaxf(mx, s);
  }
  float se = 0.f;
  for (int k = 0; k < OUTC; ++k) se += expf(lg[k] - mx);
  float lse = mx + logf(se);
  for (int k = 0; k < OUTC; ++k) out[g * OUTC + k] = lg[k] - lse;
}

// ---------------------------------------------------------------------------
extern "C" void kernel_launch(void* const* d_in, const int* in_sizes, int n_in,
                              void* d_out, int out_size, void* d_ws, size_t ws_size,
                              hipStream_t stream) {
  (void)in_sizes; (void)n_in; (void)out_size; (void)ws_size;

  const float* x      = (const float*)d_in[0];
  const int*   ei     = (const int*)d_in[1];
  const int*   batch  = (const int*)d_in[2];
  // d_in[3..6] = sa_* : softmax over a size-1 axis is identically 1 -> h == x.
  const float* g1_w   = (const float*)d_in[7];
  const float* g1_as  = (const float*)d_in[8];
  const float* g1_ad  = (const float*)d_in[9];
  const float* g1_b   = (const float*)d_in[10];
  const float* g2_w   = (const float*)d_in[11];
  const float* g2_as  = (const float*)d_in[12];
  const float* g2_ad  = (const float*)d_in[13];
  const float* g2_b   = (const float*)d_in[14];
  const float* cl_w1  = (const float*)d_in[15];
  const float* cl_b1  = (const float*)d_in[16];
  const float* cl_w2  = (const float*)d_in[17];
  const float* cl_b2  = (const float*)d_in[18];
  float* out = (float*)d_out;

  float* ws = (float*)d_ws;
  // ---- conv1 region ----
  float*    H1   = ws;                         // 50000*256 = 12.8M
  float*    ACC1 = ws + 12800000;              // 12.8M
  float*    A1S  = ws + 25600000;              // 200k
  float*    A1D  = A1S + 200000;               // 200k
  unsigned* M1   = (unsigned*)(A1D + 200000);  // 200k
  float*    D1   = (float*)M1 + 200000;        // 200k
  float*    E1   = D1 + 200000;                // 850k*4 = 3.4M
  float*    POOL = E1 + 3400000;               // 128*64
  float*    CNT  = POOL + NGRAPH * HIDC;       // 128
  // ---- conv2 region: overlays H1 space (dead after conv1 messages) ----
  float*    H2   = ws;                         // 50000*64 = 3.2M
  float*    A2S  = H2 + 3200000;
  float*    A2D  = A2S + 50000;
  unsigned* M2   = (unsigned*)(A2D + 50000);
  float*    D2   = (float*)M2 + 50000;
  float*    E2   = D2 + 50000;                 // 850k
  float*    ACC2 = E2 + 850000;                // 3.2M

  const int TB = 256;
  dim3 gemm_blk(256);
  auto cdiv = [](int a, int b) { return (a + b - 1) / b; };

  // ===== GAT layer 1 =====
  wmma_gemm_f32_kernel<<<dim3(cdiv(NNODES, 128), INCH / 64), gemm_blk, 0, stream>>>(
      x, g1_w, H1, NNODES, NHEADS * HIDC, INCH);
  att_reduce_kernel<<<cdiv(NNODES * NHEADS, TB), TB, 0, stream>>>(
      H1, g1_as, g1_ad, A1S, A1D, NHEADS, NNODES);
  fill_f32<<<cdiv(NNODES * NHEADS, TB), TB, 0, stream>>>((float*)M1, 0.f, NNODES * NHEADS);
  fill_f32<<<cdiv(NNODES * NHEADS, TB), TB, 0, stream>>>(D1, 0.f, NNODES * NHEADS);
  fill_f32<<<cdiv(NNODES * INCH, TB), TB, 0, stream>>>(ACC1, 0.f, NNODES * INCH);
  edge_max_kernel<<<cdiv(ETOT, TB), TB, 0, stream>>>(ei, A1S, A1D, M1, NHEADS);
  edge_exp_kernel<<<cdiv(ETOT, TB), TB, 0, stream>>>(ei, A1S, A1D, M1, D1, E1, NHEADS);
  edge_msg_kernel<<<cdiv(ETOT, 4), TB, 0, stream>>>(ei, E1, D1, H1, ACC1, NHEADS);
  bias_relu_kernel<<<cdiv(NNODES * INCH, TB), TB, 0, stream>>>(ACC1, g1_b, INCH, NNODES * INCH);

  // ===== GAT layer 2 =====
  wmma_gemm_f32_kernel<<<dim3(cdiv(NNODES, 128), 1), gemm_blk, 0, stream>>>(
      ACC1, g2_w, H2, NNODES, HIDC, INCH);
  att_reduce_kernel<<<cdiv(NNODES, TB), TB, 0, stream>>>(H2, g2_as, g2_ad, A2S, A2D, 1, NNODES);
  fill_f32<<<cdiv(NNODES, TB), TB, 0, stream>>>((float*)M2, 0.f, NNODES);
  fill_f32<<<cdiv(NNODES, TB), TB, 0, stream>>>(D2, 0.f, NNODES);
  fill_f32<<<cdiv(NNODES * HIDC, TB), TB, 0, stream>>>(ACC2, 0.f, NNODES * HIDC);
  fill_f32<<<cdiv(NGRAPH * HIDC, TB), TB, 0, stream>>>(POOL, 0.f, NGRAPH * HIDC);
  fill_f32<<<1, TB, 0, stream>>>(CNT, 0.f, NGRAPH);
  edge_max_kernel<<<cdiv(ETOT, TB), TB, 0, stream>>>(ei, A2S, A2D, M2, 1);
  edge_exp_kernel<<<cdiv(ETOT, TB), TB, 0, stream>>>(ei, A2S, A2D, M2, D2, E2, 1);
  edge_msg_kernel<<<cdiv(ETOT, 4), TB, 0, stream>>>(ei, E2, D2, H2, ACC2, 1);
  bias_relu_kernel<<<cdiv(NNODES * HIDC, TB), TB, 0, stream>>>(ACC2, g2_b, HIDC, NNODES * HIDC);

  // ===== pool + classifier =====
  pool_kernel<<<cdiv(NNODES * HIDC, TB), TB, 0, stream>>>(ACC2, batch, POOL, CNT);
  classifier_kernel<<<1, 128, 0, stream>>>(POOL, CNT, cl_w1, cl_b1, cl_w2, cl_b2, out);
}